// PCEN_6468220748135
// MI455X (gfx1250) — compile-verified
//
#include <hip/hip_runtime.h>

// ---------------------------------------------------------------------------
// PCEN: per-channel energy normalization.
//   ema_t = (1-s)*ema_{t-1} + s*x_t          (per (b,c) scan along T)
//   out   = (x*(FLOOR+ema)^(-alpha) + delta)^root - delta^root
//
// Strategy (bandwidth-bound, 23.3 TB/s HBM, 192MB L2 holds entire x):
//   K1: chunk-local EMA partials (L=64) -> E[b][j][c]   (2MB scratch)
//   K2: tiny serial carry scan across chunks (in-place -> carry-in per chunk)
//   K3: WMMA-based re-scan per chunk + pointwise, NT stores for output.
// The in-chunk scan in K3 is a lower-triangular 16x16 matmul done with
// V_WMMA_F32_16X16X4_F32 (4 chained K-slices), carry via accumulator init.
// ---------------------------------------------------------------------------

typedef float v2f __attribute__((ext_vector_type(2)));
typedef float v8f __attribute__((ext_vector_type(8)));

static constexpr int   Bn  = 32;
static constexpr int   Tn  = 8192;
static constexpr int   Cn  = 128;
static constexpr int   Ln  = 64;          // timesteps per chunk
static constexpr int   NC  = Tn / Ln;     // 128 chunks
static constexpr float Sc  = 0.025f;      // smooth coef
static constexpr float Ac  = 1.0f - Sc;   // 0.975
static constexpr float FLOORc = 1e-6f;

// Native transcendentals: v_log_f32 (log2) and v_exp_f32 (exp2).
__device__ __forceinline__ float fast_log2(float v) { return __builtin_amdgcn_logf(v); }
__device__ __forceinline__ float fast_exp2(float v) { return __builtin_amdgcn_exp2f(v); }
// pow(a,b) for a > 0
__device__ __forceinline__ float fast_pow(float a, float b) {
  return fast_exp2(b * fast_log2(a));
}

// ---- Kernel 1: chunk-local EMA, zero initial state --------------------------
__global__ void pcen_chunk_partial(const float* __restrict__ x,
                                   float* __restrict__ E) {
  const int j = blockIdx.x;           // chunk
  const int b = blockIdx.y;           // batch
  const int c = threadIdx.x;          // channel (128 threads, coalesced rows)
  const float* xp = x + ((size_t)b * Tn + (size_t)j * Ln) * Cn + c;
  float h = 0.0f;
#pragma unroll 8
  for (int t = 0; t < Ln; ++t) {
    h = fmaf(Ac, h, Sc * xp[(size_t)t * Cn]);
  }
  E[((size_t)b * NC + j) * Cn + c] = h;
}

// ---- Kernel 2: serial carry scan over chunks (in-place) ---------------------
//  H_j = E_j + a^L * H_{j-1};  store carry ENTERING chunk j into E[b][j][c].
__global__ void pcen_carry_scan(float* __restrict__ E) {
  const int idx = blockIdx.x * blockDim.x + threadIdx.x;   // b*Cn + c
  if (idx >= Bn * Cn) return;
  const int b = idx / Cn, c = idx % Cn;
  float aL = 1.0f;
#pragma unroll
  for (int i = 0; i < Ln; ++i) aL *= Ac;
  float cur = 0.0f;
  for (int j = 0; j < NC; ++j) {
    const size_t o = ((size_t)b * NC + j) * Cn + c;
    const float e = E[o];
    E[o] = cur;                    // carry entering chunk j
    cur = fmaf(aL, cur, e);
  }
}

// ---- Kernel 3: WMMA tile scan + pointwise PCEN ------------------------------
// 256 threads = 8 waves; wave w owns channels [16w, 16w+16).
// Per 16-timestep tile:  H(16x16) = M(16x16, lower-tri s*a^(i-u)) * X(16x16)
//                                   + Cinit[m][n] = a^(m+1)*h_in[n]
// done as 4 chained V_WMMA_F32_16X16X4_F32.
__global__ void pcen_apply_wmma(const float* __restrict__ x,
                                const float* __restrict__ alpha,
                                const float* __restrict__ delta,
                                const float* __restrict__ root,
                                const float* __restrict__ Cin,
                                float* __restrict__ out) {
  const int j    = blockIdx.x;
  const int b    = blockIdx.y;
  const int tid  = threadIdx.x;
  const int wave = tid >> 5;
  const int lane = tid & 31;
  const int half = lane >> 4;       // 0: lanes 0-15, 1: lanes 16-31
  const int n    = lane & 15;       // column / channel-in-tile
  const int ch   = wave * 16 + n;   // global channel

  // Per-channel constants.
  const float na = -alpha[ch];
  const float dl = delta[ch];
  const float rt = root[ch];
  const float dr = fast_pow(dl, rt);   // delta^root

  // A-matrix K-slices: lane holds row i = n; reg jj holds u = 4k + 2*half + jj.
  const int i = n;
  v2f mA[4];
#pragma unroll
  for (int k = 0; k < 4; ++k) {
#pragma unroll
    for (int jj = 0; jj < 2; ++jj) {
      const int u = 4 * k + 2 * half + jj;
      float v = 0.0f;
      if (u <= i) {
        float p = Sc;
        for (int d = 0; d < i - u; ++d) p *= Ac;
        v = p;
      }
      mA[k][jj] = v;
    }
  }

  // Carry coefficients for accumulator init: rows m = v + 8*half -> a^(m+1).
  float apow[8];
#pragma unroll
  for (int v = 0; v < 8; ++v) {
    const int m = v + 8 * half;
    float p = Ac;
    for (int d = 0; d < m; ++d) p *= Ac;
    apow[v] = p;
  }

  float h_in = Cin[((size_t)b * NC + j) * Cn + ch];   // carry entering chunk
  const size_t xbase = ((size_t)b * Tn + (size_t)j * Ln) * Cn;

  for (int tt = 0; tt < Ln / 16; ++tt) {
    const int t0 = tt * 16;

    // Accumulator init = carry outer product.
    v8f acc;
#pragma unroll
    for (int v = 0; v < 8; ++v) acc[v] = apow[v] * h_in;

    // 4 K-slices of the 16x16 product.
#pragma unroll
    for (int k = 0; k < 4; ++k) {
      v2f bx;
#pragma unroll
      for (int jj = 0; jj < 2; ++jj) {
        const int t = t0 + 4 * k + 2 * half + jj;   // B: K = 2*half + jj
        bx[jj] = x[xbase + (size_t)t * Cn + ch];
      }
      acc = __builtin_amdgcn_wmma_f32_16x16x4_f32(
          /*neg_a=*/false, mA[k], /*neg_b=*/false, bx,
          /*c_mod=*/(short)0, acc, /*reuse_a=*/false, /*reuse_b=*/false);
    }

    // Carry to next tile: D row 15 lives in VGPR7 of lanes 16..31.
    h_in = __shfl(acc[7], 16 + n, 32);

    // Pointwise PCEN + non-temporal store (keep x resident in L2).
#pragma unroll
    for (int v = 0; v < 8; ++v) {
      const int t = t0 + v + 8 * half;              // D row m = v + 8*half
      const size_t o = xbase + (size_t)t * Cn + ch;
      const float xv  = x[o];                        // L0/L2 hit (just read)
      const float ema = acc[v];
      const float sm  = fast_exp2(na * fast_log2(FLOORc + ema));
      const float res = fast_exp2(rt * fast_log2(fmaf(xv, sm, dl))) - dr;
      __builtin_nontemporal_store(res, out + o);
    }
  }
}

// ---------------------------------------------------------------------------
extern "C" void kernel_launch(void* const* d_in, const int* in_sizes, int n_in,
                              void* d_out, int out_size, void* d_ws, size_t ws_size,
                              hipStream_t stream) {
  (void)in_sizes; (void)n_in; (void)out_size; (void)ws_size;
  const float* x     = (const float*)d_in[0];
  const float* alpha = (const float*)d_in[1];
  const float* delta = (const float*)d_in[2];
  const float* root  = (const float*)d_in[3];
  float* out = (float*)d_out;
  float* E   = (float*)d_ws;   // NC*Bn*Cn floats = 2 MB scratch

  pcen_chunk_partial<<<dim3(NC, Bn), 128, 0, stream>>>(x, E);
  pcen_carry_scan<<<dim3((Bn * Cn + 255) / 256), 256, 0, stream>>>(E);
  pcen_apply_wmma<<<dim3(NC, Bn), 256, 0, stream>>>(x, alpha, delta, root, E, out);
}